// GraphConvolutionTopK_17824114278496
// MI455X (gfx1250) — compile-verified
//
#include <hip/hip_runtime.h>
#include <cstdint>
#include <cstddef>

typedef float v2f __attribute__((ext_vector_type(2)));
typedef float v8f __attribute__((ext_vector_type(8)));

#define BB     8
#define CC     256      // C_in == C_out
#define NN     2048
#define K_TOP  1843     // round(2048 * 0.9)
#define LEAKY  0.01f
#define BN_EPS 1e-5f

// ---------------------------------------------------------------------------
// K1: gram logits  S[b,i,j] = sum_c x[b,c,i] * x[b,c,j]
// one wave per 16x128 tile strip: 8 accumulators share each A fragment.
// ---------------------------------------------------------------------------
__global__ __launch_bounds__(32) void k_gram(const float* __restrict__ x,
                                             float* __restrict__ adj) {
  const int b    = blockIdx.y;
  const int ti   = blockIdx.x / (NN / 128);   // 16-row tile index
  const int tj   = blockIdx.x % (NN / 128);   // 128-col block index
  const int lane = threadIdx.x & 31;
  const int lo   = lane & 15;
  const int hi   = lane >> 4;

  const float* xb = x + (size_t)b * CC * NN;
  const float* pa = xb + (size_t)(2 * hi) * NN + ti * 16 + lo;   // A: M=lo
  const float* pb = xb + (size_t)(2 * hi) * NN + tj * 128 + lo;  // B: N=lo (+u*16)

  v8f acc[8];
#pragma unroll
  for (int u = 0; u < 8; ++u) acc[u] = (v8f){};

  for (int c0 = 0; c0 < CC; c0 += 4) {
    v2f a;
    a.x = pa[0];
    a.y = pa[NN];
#pragma unroll
    for (int u = 0; u < 8; ++u) {
      v2f bm;
      bm.x = pb[u * 16];
      bm.y = pb[NN + u * 16];
      acc[u] = __builtin_amdgcn_wmma_f32_16x16x4_f32(false, a, false, bm,
                                                     (short)0, acc[u], false, false);
    }
    pa += 4 * (size_t)NN;
    pb += 4 * (size_t)NN;
  }

#pragma unroll
  for (int u = 0; u < 8; ++u) {
    float* out = adj + ((size_t)b * NN + ti * 16) * NN + tj * 128 + u * 16;
#pragma unroll
    for (int r = 0; r < 8; ++r)
      out[(size_t)(r + 8 * hi) * NN + lo] = acc[u][r];
  }
}

// ---------------------------------------------------------------------------
// K2: row softmax over adj (in place).  One block (256 thr) per row of 2048.
// ---------------------------------------------------------------------------
__global__ __launch_bounds__(256) void k_softmax(float* __restrict__ adj) {
  const int row = blockIdx.x;
  float* p = adj + (size_t)row * NN;
  __shared__ float red[256];
  const int t = threadIdx.x;

  float v[8];
  float vmax = -3.402823466e38f;
#pragma unroll
  for (int i = 0; i < 8; ++i) { v[i] = p[t + 256 * i]; vmax = fmaxf(vmax, v[i]); }
  red[t] = vmax; __syncthreads();
  for (int s = 128; s > 0; s >>= 1) {
    if (t < s) red[t] = fmaxf(red[t], red[t + s]);
    __syncthreads();
  }
  vmax = red[0]; __syncthreads();

  float sum = 0.f;
#pragma unroll
  for (int i = 0; i < 8; ++i) { v[i] = __expf(v[i] - vmax); sum += v[i]; }
  red[t] = sum; __syncthreads();
  for (int s = 128; s > 0; s >>= 1) {
    if (t < s) red[t] += red[t + s];
    __syncthreads();
  }
  const float inv = 1.0f / red[0];
#pragma unroll
  for (int i = 0; i < 8; ++i) p[t + 256 * i] = v[i] * inv;
}

// ---------------------------------------------------------------------------
// K3: exact k-th largest per row via 4-pass radix select on float bits
// (softmax outputs are positive => bit pattern order == value order)
// ---------------------------------------------------------------------------
__global__ __launch_bounds__(256) void k_topk(const float* __restrict__ adj,
                                              float* __restrict__ thresh) {
  const int row = blockIdx.x;
  const uint32_t* p = (const uint32_t*)(adj + (size_t)row * NN);
  __shared__ uint32_t bits[NN];
  __shared__ uint32_t hist[256];
  __shared__ uint32_t s_prefix, s_remaining;
  const int t = threadIdx.x;

#pragma unroll
  for (int i = 0; i < 8; ++i) bits[t + 256 * i] = p[t + 256 * i];
  if (t == 0) { s_prefix = 0u; s_remaining = K_TOP; }
  __syncthreads();

  for (int shift = 24; shift >= 0; shift -= 8) {
    hist[t] = 0u; __syncthreads();
    const uint32_t pref = s_prefix;
    const uint32_t mask = (shift == 24) ? 0u : (0xFFFFFFFFu << (shift + 8));
#pragma unroll
    for (int i = 0; i < 8; ++i) {
      const uint32_t vb = bits[t + 256 * i];
      if ((vb & mask) == pref) atomicAdd(&hist[(vb >> shift) & 0xFF], 1u);
    }
    __syncthreads();
    if (t == 0) {
      uint32_t rem = s_remaining, cum = 0;
      for (int bkt = 255; bkt >= 0; --bkt) {
        cum += hist[bkt];
        if (cum >= rem) {
          s_remaining = rem - (cum - hist[bkt]);
          s_prefix    = pref | ((uint32_t)bkt << shift);
          break;
        }
      }
    }
    __syncthreads();
  }
  if (t == 0) thresh[row] = __uint_as_float(s_prefix);
}

// ---------------------------------------------------------------------------
// K4: support[b,n,d] = sum_c x[b,c,n] * w[c,d]
// one wave per 16x64 tile strip: 4 accumulators share each A fragment.
// ---------------------------------------------------------------------------
__global__ __launch_bounds__(32) void k_support(const float* __restrict__ x,
                                                const float* __restrict__ w,
                                                float* __restrict__ sup) {
  const int b    = blockIdx.y;
  const int ti   = blockIdx.x / (CC / 64);
  const int td   = blockIdx.x % (CC / 64);
  const int lane = threadIdx.x & 31;
  const int lo   = lane & 15;
  const int hi   = lane >> 4;

  const float* pa = x + (size_t)b * CC * NN + (size_t)(2 * hi) * NN + ti * 16 + lo;
  const float* pb = w + (size_t)(2 * hi) * CC + td * 64 + lo;

  v8f acc[4];
#pragma unroll
  for (int u = 0; u < 4; ++u) acc[u] = (v8f){};

  for (int c0 = 0; c0 < CC; c0 += 4) {
    v2f a;
    a.x = pa[0];
    a.y = pa[NN];
#pragma unroll
    for (int u = 0; u < 4; ++u) {
      v2f bm;
      bm.x = pb[u * 16];
      bm.y = pb[CC + u * 16];
      acc[u] = __builtin_amdgcn_wmma_f32_16x16x4_f32(false, a, false, bm,
                                                     (short)0, acc[u], false, false);
    }
    pa += 4 * (size_t)NN;
    pb += 4 * (size_t)CC;
  }

#pragma unroll
  for (int u = 0; u < 4; ++u) {
    float* out = sup + ((size_t)b * NN + ti * 16) * CC + td * 64 + u * 16;
#pragma unroll
    for (int r = 0; r < 8; ++r)
      out[(size_t)(r + 8 * hi) * CC + lo] = acc[u][r];
  }
}

// ---------------------------------------------------------------------------
// K5: y[b,n,d] = leaky( sum_m mask(adj[b,n,m]) * sup[b,m,d] )
// one wave per 16x64 strip: masked A fragment loaded once per 4 WMMAs,
// cutting total adjacency traffic 4x vs per-tile waves.
// ---------------------------------------------------------------------------
__global__ __launch_bounds__(32) void k_out(const float* __restrict__ adj,
                                            const float* __restrict__ thresh,
                                            const float* __restrict__ sup,
                                            float* __restrict__ y) {
  const int b    = blockIdx.y;
  const int ti   = blockIdx.x / (CC / 64);
  const int td   = blockIdx.x % (CC / 64);
  const int lane = threadIdx.x & 31;
  const int lo   = lane & 15;
  const int hi   = lane >> 4;

  const float th  = thresh[b * NN + ti * 16 + lo];
  const float* pa = adj + ((size_t)b * NN + ti * 16 + lo) * NN + 2 * hi;
  const float* pb = sup + ((size_t)b * NN + 2 * hi) * CC + td * 64 + lo;

  v8f acc[4];
#pragma unroll
  for (int u = 0; u < 4; ++u) acc[u] = (v8f){};

  for (int m0 = 0; m0 < NN; m0 += 4) {
    const float a0 = pa[0], a1 = pa[1];
    v2f a;
    a.x = (a0 >= th) ? a0 : 0.0f;
    a.y = (a1 >= th) ? a1 : 0.0f;
#pragma unroll
    for (int u = 0; u < 4; ++u) {
      v2f bm;
      bm.x = pb[u * 16];
      bm.y = pb[CC + u * 16];
      acc[u] = __builtin_amdgcn_wmma_f32_16x16x4_f32(false, a, false, bm,
                                                     (short)0, acc[u], false, false);
    }
    pa += 4;
    pb += 4 * (size_t)CC;
  }

#pragma unroll
  for (int u = 0; u < 4; ++u) {
    float* out = y + ((size_t)b * NN + ti * 16) * CC + td * 64 + u * 16;
#pragma unroll
    for (int r = 0; r < 8; ++r) {
      float v = acc[u][r];
      v = (v >= 0.0f) ? v : LEAKY * v;
      out[(size_t)(r + 8 * hi) * CC + lo] = v;
    }
  }
}

// ---------------------------------------------------------------------------
// K6: batchnorm stats (sum, sumsq per channel) then normalize + transpose
// ---------------------------------------------------------------------------
__global__ __launch_bounds__(256) void k_stats_init(float* __restrict__ stats) {
  stats[blockIdx.x * 256 + threadIdx.x] = 0.0f;
}

__global__ __launch_bounds__(256) void k_stats(const float* __restrict__ y,
                                               float* __restrict__ stats) {
  const int t  = threadIdx.x;          // channel d == t
  const int r0 = blockIdx.x * 256;     // 64 blocks x 256 rows = 16384 rows
  float s = 0.f, s2 = 0.f;
  for (int r = 0; r < 256; ++r) {
    const float v = y[(size_t)(r0 + r) * CC + t];
    s  += v;
    s2 += v * v;
  }
  atomicAdd(&stats[t], s);
  atomicAdd(&stats[CC + t], s2);
}

__global__ __launch_bounds__(256) void k_bn(const float* __restrict__ y,
                                            const float* __restrict__ stats,
                                            const float* __restrict__ gamma,
                                            const float* __restrict__ beta,
                                            float* __restrict__ out) {
  const int b = blockIdx.y;
  const int d = blockIdx.x >> 3;
  const int n = (blockIdx.x & 7) * 256 + threadIdx.x;
  const float cnt  = (float)(BB * NN);
  const float mean = stats[d] / cnt;
  const float var  = stats[CC + d] / cnt - mean * mean;
  float v = y[((size_t)b * NN + n) * CC + d];
  v = (v - mean) * rsqrtf(var + BN_EPS);
  out[((size_t)b * CC + d) * NN + n] = v * gamma[d] + beta[d];
}

// ---------------------------------------------------------------------------
extern "C" void kernel_launch(void* const* d_in, const int* in_sizes, int n_in,
                              void* d_out, int out_size, void* d_ws, size_t ws_size,
                              hipStream_t stream) {
  const float* x      = (const float*)d_in[0];
  const float* weight = (const float*)d_in[1];
  const float* bn_w   = (const float*)d_in[2];
  const float* bn_b   = (const float*)d_in[3];
  float* out = (float*)d_out;

  char* ws = (char*)d_ws;
  float* adj    = (float*)ws;                                              // 134217728 B
  float* sup    = (float*)(ws + (size_t)134217728);                        // 16777216 B
  float* y      = (float*)(ws + (size_t)134217728 + 16777216);             // 16777216 B
  float* thresh = (float*)(ws + (size_t)134217728 + 2 * 16777216);         // 65536 B
  float* stats  = (float*)(ws + (size_t)134217728 + 2 * 16777216 + 65536); // 2048 B

  // 1. gram logits (16x128 strips per wave)
  k_gram<<<dim3((NN / 16) * (NN / 128), BB), 32, 0, stream>>>(x, adj);
  // 2. row softmax
  k_softmax<<<BB * NN, 256, 0, stream>>>(adj);
  // 3. per-row top-k threshold (exact radix select)
  k_topk<<<BB * NN, 256, 0, stream>>>(adj, thresh);
  // 4. support = xn @ W (16x64 strips per wave)
  k_support<<<dim3((NN / 16) * (CC / 64), BB), 32, 0, stream>>>(x, weight, sup);
  // 5. masked adj @ support + leaky relu (16x64 strips per wave)
  k_out<<<dim3((NN / 16) * (CC / 64), BB), 32, 0, stream>>>(adj, thresh, sup, y);
  // 6. batchnorm
  k_stats_init<<<2, 256, 0, stream>>>(stats);
  k_stats<<<64, 256, 0, stream>>>(y, stats);
  k_bn<<<dim3((CC) * (NN / 256), BB), 256, 0, stream>>>(y, stats, bn_w, bn_b, out);
}